// DeformableAttention_13924283974145
// MI455X (gfx1250) — compile-verified
//
#include <hip/hip_runtime.h>
#include <math.h>

typedef __attribute__((ext_vector_type(2))) float v2f;
typedef __attribute__((ext_vector_type(8))) float v8f;

#define NHEAD 8
#define NPNT 4
#define HDIM 32
#define HSP 128
#define WSP 128
#define NQ 16384
#define BATCH 2
#define MTOT (BATCH * NQ)   // 32768

// ---------------------------------------------------------------------------
// fp32 GEMM: C[M x N] = A[M x 256] * W[256 x N] + bias[N]
// Block = 256 threads = 8 waves; block covers 128 rows x (NT*16) cols.
// W column strip is staged in LDS (two 128-row chunks) and shared by all 8
// waves; each wave computes NT 16x16 tiles via V_WMMA_F32_16X16X4_F32 so one
// A fragment feeds NT WMMAs.
//   A frag (16x4): lane l holds A[m=l&15][k + 2*(l>>4) + {0,1}]
//   B frag (4x16): lane l holds W[k + 2*(l>>4) + {0,1}][n=l&15]
//   C/D (16x16):   vgpr r, lane l -> M = r + 8*(l>>4), N = l&15
// LDS row stride = COLS+8 so rows k and k+2 (the two lane-halves) map to
// disjoint 16-bank groups (2*(COLS+8) == 16 mod 64).
// NOTE: requires M % 128 == 0 (no early-exit; all waves reach barriers).
// ---------------------------------------------------------------------------
template <int NT>
__global__ __launch_bounds__(256) void gemm_k256(
    const float* __restrict__ A, const float* __restrict__ W,
    const float* __restrict__ bias, float* __restrict__ C, int M, int N) {
  constexpr int COLS = NT * 16;
  constexpr int LSTR = COLS + 8;          // padded LDS row stride (floats)
  __shared__ float sW[128 * LSTR];

  const int wave = threadIdx.x >> 5;
  const int lane = threadIdx.x & 31;
  const int m0 = blockIdx.x * 128 + wave * 16;
  const int n0 = blockIdx.y * COLS;
  const int half = lane >> 4;             // 0 or 1
  const int l15 = lane & 15;
  const int row = m0 + l15;

  const float* __restrict__ Arow = A + (size_t)row * 256 + half * 2;

  v8f c[NT];
  #pragma unroll
  for (int t = 0; t < NT; ++t) c[t] = (v8f){};

  for (int kc = 0; kc < 256; kc += 128) {
    // cooperative, coalesced stage of 128 rows of the W column strip
    for (int i = threadIdx.x; i < 128 * COLS; i += 256) {
      const int kr = i / COLS;
      const int cc = i - kr * COLS;
      sW[kr * LSTR + cc] = W[(size_t)(kc + kr) * N + n0 + cc];
    }
    __syncthreads();

    if (kc == 0) __builtin_prefetch(Arow + 128, 0, 1);  // next A chunk

    #pragma unroll 4
    for (int k = 0; k < 128; k += 4) {
      v2f a;
      a.x = Arow[kc + k];
      a.y = Arow[kc + k + 1];
      const int r0 = (k + 2 * half) * LSTR + l15;
      #pragma unroll
      for (int t = 0; t < NT; ++t) {
        v2f b;
        b.x = sW[r0 + t * 16];
        b.y = sW[r0 + LSTR + t * 16];
        c[t] = __builtin_amdgcn_wmma_f32_16x16x4_f32(
            /*neg_a=*/false, a, /*neg_b=*/false, b,
            /*c_mod=*/(short)0, c[t], /*reuse_a=*/false, /*reuse_b=*/false);
      }
    }
    __syncthreads();
  }

  #pragma unroll
  for (int t = 0; t < NT; ++t) {
    const int col = n0 + t * 16 + l15;
    const float bv = bias[col];
    #pragma unroll
    for (int r = 0; r < 8; ++r) {
      const int orow = m0 + r + 8 * half;
      C[(size_t)orow * N + col] = c[t][r] + bv;
    }
  }
}

// ---------------------------------------------------------------------------
// Deformable sampling. One wave per query row.
// Phase 1: lane == (h,p) index (h = lane>>2, p = lane&3): tanh of offsets,
//          softmax over the 4 points of each head (shfl_xor within 4 lanes).
// Phase 2: lane == head-dim element; loop hp=0..31, broadcast sample coords
//          via shfl, gather 4 bilinear taps (coalesced 128B lines from vbuf),
//          accumulate attn-weighted result per head.
// ---------------------------------------------------------------------------
__global__ __launch_bounds__(256) void deform_sample(
    const float* __restrict__ vbuf,     // (B, H, W, NH, HD) == (B*NQ, 256)
    const float* __restrict__ offraw,   // (B*NQ, 64): (h,p,{x,y})
    const float* __restrict__ attnraw,  // (B*NQ, 32): (h,p)
    float* __restrict__ mid)            // (B*NQ, 256)
{
  const int wave = threadIdx.x >> 5;
  const int lane = threadIdx.x & 31;
  const int gq = blockIdx.x * 8 + wave;       // row in [0, B*NQ)
  if (gq >= MTOT) return;
  const int b  = gq / NQ;
  const int q  = gq - b * NQ;
  const int qy = q >> 7;                      // q / 128
  const int qx = q & 127;

  // --- per-(h,p) post-processing on lane==hp ---
  const float ox    = tanhf(offraw[(size_t)gq * 64 + 2 * lane]);
  const float oy    = tanhf(offraw[(size_t)gq * 64 + 2 * lane + 1]);
  const float logit = attnraw[(size_t)gq * 32 + lane];

  float mx = logit;
  mx = fmaxf(mx, __shfl_xor(mx, 1, 32));
  mx = fmaxf(mx, __shfl_xor(mx, 2, 32));
  float e = __expf(logit - mx);
  float s = e;
  s += __shfl_xor(s, 1, 32);
  s += __shfl_xor(s, 2, 32);
  const float aw = e / s;

  const float refx = -1.0f + 2.0f * (float)qx / 127.0f;
  const float refy = -1.0f + 2.0f * (float)qy / 127.0f;
  const float locx = fminf(fmaxf(refx + ox * 0.5f, -1.0f), 1.0f);
  const float locy = fminf(fmaxf(refy + oy * 0.5f, -1.0f), 1.0f);
  const float ixl  = ((locx + 1.0f) * (float)WSP - 1.0f) * 0.5f;
  const float iyl  = ((locy + 1.0f) * (float)HSP - 1.0f) * 0.5f;

  // --- gather phase: lane == head-dim element ---
  float acc[NHEAD];
  #pragma unroll
  for (int h = 0; h < NHEAD; ++h) acc[h] = 0.0f;

  const float* __restrict__ vb = vbuf + (size_t)b * NQ * 256;

  for (int hp = 0; hp < 32; ++hp) {
    const float sx = __shfl(ixl, hp, 32);
    const float sy = __shfl(iyl, hp, 32);
    const float sw = __shfl(aw,  hp, 32);
    const int h = hp >> 2;

    const float x0f = floorf(sx), y0f = floorf(sy);
    const int x0 = (int)x0f, y0 = (int)y0f;
    const int x1 = x0 + 1,   y1 = y0 + 1;
    const float wx1 = sx - x0f, wx0 = 1.0f - wx1;
    const float wy1 = sy - y0f, wy0 = 1.0f - wy1;

    const bool vx0 = (x0 >= 0) && (x0 < WSP);
    const bool vx1 = (x1 >= 0) && (x1 < WSP);
    const bool vy0 = (y0 >= 0) && (y0 < HSP);
    const bool vy1 = (y1 >= 0) && (y1 < HSP);
    const int cx0 = min(max(x0, 0), WSP - 1);
    const int cx1 = min(max(x1, 0), WSP - 1);
    const int cy0 = min(max(y0, 0), HSP - 1);
    const int cy1 = min(max(y1, 0), HSP - 1);

    const int hd = h * HDIM + lane;
    float v00 = 0.f, v10 = 0.f, v01 = 0.f, v11 = 0.f;
    if (vy0) {                                   // uniform branches
      const float* rowp = vb + (size_t)(cy0 * WSP) * 256 + hd;
      if (vx0) v00 = rowp[(size_t)cx0 * 256];
      if (vx1) v10 = rowp[(size_t)cx1 * 256];
    }
    if (vy1) {
      const float* rowp = vb + (size_t)(cy1 * WSP) * 256 + hd;
      if (vx0) v01 = rowp[(size_t)cx0 * 256];
      if (vx1) v11 = rowp[(size_t)cx1 * 256];
    }
    acc[h] += sw * (wx0 * wy0 * v00 + wx1 * wy0 * v10 +
                    wx0 * wy1 * v01 + wx1 * wy1 * v11);
  }

  float* __restrict__ op = mid + (size_t)gq * 256 + lane;
  #pragma unroll
  for (int h = 0; h < NHEAD; ++h) op[(size_t)h * HDIM] = acc[h];
}

// ---------------------------------------------------------------------------
extern "C" void kernel_launch(void* const* d_in, const int* in_sizes, int n_in,
                              void* d_out, int out_size, void* d_ws, size_t ws_size,
                              hipStream_t stream) {
  const float* query  = (const float*)d_in[0];
  const float* value  = (const float*)d_in[1];
  const float* W_off  = (const float*)d_in[2];
  const float* b_off  = (const float*)d_in[3];
  const float* W_attn = (const float*)d_in[4];
  const float* b_attn = (const float*)d_in[5];
  const float* W_v    = (const float*)d_in[6];
  const float* b_v    = (const float*)d_in[7];
  const float* W_out  = (const float*)d_in[8];
  const float* b_out  = (const float*)d_in[9];
  // d_in[10], d_in[11] are H=128, W=128 scalars (compile-time constants here)
  float* out = (float*)d_out;

  char* ws = (char*)d_ws;
  float* vbuf    = (float*)(ws);                               // 32 MB
  float* mid     = (float*)(ws + (size_t)32 * 1024 * 1024);    // 32 MB
  float* offraw  = (float*)(ws + (size_t)64 * 1024 * 1024);    //  8 MB
  float* attnraw = (float*)(ws + (size_t)72 * 1024 * 1024);    //  4 MB

  const dim3 blk(256);

  // 1) v = value @ W_v + b_v           (M=32768, N=256)
  gemm_k256<4><<<dim3(MTOT / 128, 256 / 64), blk, 0, stream>>>(
      value, W_v, b_v, vbuf, MTOT, 256);
  // 2) offraw = query @ W_off + b_off  (N=64 -> one 4-tile strip)
  gemm_k256<4><<<dim3(MTOT / 128, 1), blk, 0, stream>>>(
      query, W_off, b_off, offraw, MTOT, 64);
  // 3) attnraw = query @ W_attn + b_attn (N=32 -> one 2-tile strip)
  gemm_k256<2><<<dim3(MTOT / 128, 1), blk, 0, stream>>>(
      query, W_attn, b_attn, attnraw, MTOT, 32);
  // 4) bilinear deformable sampling -> mid
  deform_sample<<<dim3(MTOT / 8), blk, 0, stream>>>(vbuf, offraw, attnraw, mid);
  // 5) out = mid @ W_out + b_out       (M=32768, N=256)
  gemm_k256<4><<<dim3(MTOT / 128, 256 / 64), blk, 0, stream>>>(
      mid, W_out, b_out, out, MTOT, 256);
}